// KroneckerRotationAttention_10307921510520
// MI455X (gfx1250) — compile-verified
//
#include <hip/hip_runtime.h>

// ---------------------------------------------------------------------------
// KroneckerRotationAttention for MI455X (gfx1250), fp32 via V_WMMA_F32_16X16X4_F32
// with double-buffered async global->LDS tile pipelines (ASYNCcnt).
// B=2, T=2048, C=1024, H=16, DK=64, DV=176, DVP=256, OFFSET=64, L=2112
// d_out = [ y (B,T,C) | k (B,H,L,DK) | v (B,H,L,DVP) ]  (fp32, concatenated)
// d_ws  = [ q (B,H,T,DK) 16MB | rot_per_head (B,H,T,DV) 46MB ]
// ---------------------------------------------------------------------------

typedef __attribute__((ext_vector_type(2))) float v2f;
typedef __attribute__((ext_vector_type(8))) float v8f;
typedef int v4i __attribute__((vector_size(16)));

#define Hh   16
#define DK   64
#define DV   176
#define DVP  256
#define Bb   2
#define Tt   2048
#define Cc   1024
#define OFF  64
#define LL   2112

__device__ __forceinline__ v8f wmma4(v2f a, v2f b, v8f c) {
  // D = A(16x4) * B(4x16) + C(16x16), fp32
  return __builtin_amdgcn_wmma_f32_16x16x4_f32(false, a, false, b, (short)0, c,
                                               false, false);
}

// ---- async global->LDS copy (CDNA5 GLOBAL_LOAD_ASYNC_TO_LDS_B128) ----------
#if defined(__has_builtin)
#if __has_builtin(__builtin_amdgcn_global_load_async_to_lds_b128)
#define HAVE_ASYNC_LDS 1
#endif
#endif
#ifndef HAVE_ASYNC_LDS
#define HAVE_ASYNC_LDS 0
#endif

__device__ __forceinline__ void copy16_g2l(const float* g, float* l) {
#if HAVE_ASYNC_LDS
  __builtin_amdgcn_global_load_async_to_lds_b128(
      (__attribute__((address_space(1))) v4i*)(uintptr_t)g,
      (__attribute__((address_space(3))) v4i*)(uint32_t)(uintptr_t)l,
      0, 0);
#else
  *(float4*)l = *(const float4*)g;
#endif
}

__device__ __forceinline__ void async_join() {
#if HAVE_ASYNC_LDS
#if defined(__has_builtin) && __has_builtin(__builtin_amdgcn_s_wait_asynccnt)
  __builtin_amdgcn_s_wait_asynccnt(0);
#else
  asm volatile("s_wait_asynccnt 0x0" ::: "memory");
#endif
#endif
}

// ============================================================================
// Kernel 1: tiled GEMM  C[m,n] = sum_k X[m,k] * W[n,k]   (M=4096, K=1024)
// mode 0: W_QK (N=2048) -> q to ws, k to kout   mode 1: W_V (N=2816) -> vout
// Block: 256 thr (8 waves). Tile 128x64. Wave = 32x32 (2x2 of 16x16 WMMA).
// Double-buffered async global->LDS pipeline over K.
// ============================================================================
#define AST 36  // LDS row stride (dwords): 16B aligned, conflict-free
__global__ __launch_bounds__(256) void gemm_proj_kernel(
    const float* __restrict__ X, const float* __restrict__ W, int mode,
    float* __restrict__ qws, float* __restrict__ kout, float* __restrict__ vout) {
  __shared__ __align__(16) float Al[2][128 * AST];
  __shared__ __align__(16) float Bl[2][64 * AST];

  const int m0 = blockIdx.x * 128;
  const int n0 = blockIdx.y * 64;
  const int tid = threadIdx.x;
  const int wave = tid >> 5, lane = tid & 31;
  const int wr = wave >> 1, wc = wave & 1;     // wave grid 4x2
  const int half = lane >> 4, ln = lane & 15;

  auto load_tiles = [&](int k0, int buf) {
    // A tile 128x32 : 1024 x b128 -> 4 per thread
#pragma unroll
    for (int j = 0; j < 4; j++) {
      int v = tid + j * 256;
      int row = v >> 3, kq = (v & 7) * 4;
      copy16_g2l(X + (size_t)(m0 + row) * Cc + k0 + kq, &Al[buf][row * AST + kq]);
    }
    // B tile 64x32 : 512 x b128 -> 2 per thread
#pragma unroll
    for (int j = 0; j < 2; j++) {
      int v = tid + j * 256;
      int row = v >> 3, kq = (v & 7) * 4;
      copy16_g2l(W + (size_t)(n0 + row) * Cc + k0 + kq, &Bl[buf][row * AST + kq]);
    }
  };

  v8f acc[2][2];
#pragma unroll
  for (int i = 0; i < 2; i++)
#pragma unroll
    for (int j = 0; j < 2; j++)
      acc[i][j] = (v8f){0.f, 0.f, 0.f, 0.f, 0.f, 0.f, 0.f, 0.f};

  load_tiles(0, 0);
  async_join();
  __syncthreads();

  for (int k0 = 0; k0 < Cc; k0 += 32) {
    const int cur = (k0 >> 5) & 1;
    if (k0 + 32 < Cc) load_tiles(k0 + 32, cur ^ 1);  // prefetch next tile

    const float* Ab = Al[cur];
    const float* Bbuf = Bl[cur];
#pragma unroll
    for (int s = 0; s < 8; s++) {
      const int kk = 4 * s + half * 2;
      v2f a0, a1, b0, b1;
      a0.x = Ab[(wr * 32 + ln) * AST + kk];
      a0.y = Ab[(wr * 32 + ln) * AST + kk + 1];
      a1.x = Ab[(wr * 32 + 16 + ln) * AST + kk];
      a1.y = Ab[(wr * 32 + 16 + ln) * AST + kk + 1];
      b0.x = Bbuf[(wc * 32 + ln) * AST + kk];
      b0.y = Bbuf[(wc * 32 + ln) * AST + kk + 1];
      b1.x = Bbuf[(wc * 32 + 16 + ln) * AST + kk];
      b1.y = Bbuf[(wc * 32 + 16 + ln) * AST + kk + 1];
      acc[0][0] = wmma4(a0, b0, acc[0][0]);
      acc[0][1] = wmma4(a0, b1, acc[0][1]);
      acc[1][0] = wmma4(a1, b0, acc[1][0]);
      acc[1][1] = wmma4(a1, b1, acc[1][1]);
    }
    async_join();
    __syncthreads();
  }

  // scatter-store
#pragma unroll
  for (int mi = 0; mi < 2; mi++)
#pragma unroll
    for (int ni = 0; ni < 2; ni++)
#pragma unroll
      for (int vr = 0; vr < 8; vr++) {
        int row = m0 + wr * 32 + mi * 16 + vr + half * 8;  // 0..4095
        int col = n0 + wc * 32 + ni * 16 + ln;
        float val = acc[mi][ni][vr];
        int b = row >> 11, t = row & 2047;
        if (mode == 0) {
          if (col < 1024) {  // q
            int h = col >> 6, d = col & 63;
            qws[(((size_t)b * Hh + h) * Tt + t) * DK + d] = val;
          } else {           // k
            int c2 = col - 1024;
            int h = c2 >> 6, d = c2 & 63;
            kout[(((size_t)b * Hh + h) * LL + OFF + t) * DK + d] = val;
          }
        } else {             // v (unpadded cols only; pad zeroed elsewhere)
          int h = col / DV, d = col - h * DV;
          vout[(((size_t)b * Hh + h) * LL + OFF + t) * DVP + d] = val;
        }
      }
}

// ============================================================================
// Kernel 2: flash attention. Block = (b*H+h, 64-query tile), 256 thr / 8 waves.
// Wave (rg,ch): rg = 16-query row group (0..3), ch = 128-col half of V (0..1).
// Key tiles of 16, double-buffered async K/V loads. Online softmax in WMMA
// C-layout; P via per-wave LDS -> A-layout.
// ============================================================================
#define KT 16
#define KS 68   // K LDS stride
#define VS 260  // V LDS stride
#define PS 20   // P LDS stride
__global__ __launch_bounds__(256) void attn_kernel(
    const float* __restrict__ qws, const float* __restrict__ kout,
    const float* __restrict__ vout, float* __restrict__ rotheads) {
  __shared__ __align__(16) float Kl[2][KT * KS];
  __shared__ __align__(16) float Vl[2][KT * VS];
  __shared__ __align__(16) float Pl[8 * 16 * PS];

  const int qtile = blockIdx.x & 31;
  const int bh = blockIdx.x >> 5;      // b*H+h
  const int q0 = qtile * 64;
  const int tid = threadIdx.x;
  const int wave = tid >> 5, lane = tid & 31;
  const int rg = wave >> 1, ch = wave & 1;
  const int half = lane >> 4, ln = lane & 15;

  const float* Kg = kout + (size_t)bh * LL * DK;
  const float* Vg = vout + (size_t)bh * LL * DVP;
  float* Pw = Pl + wave * 16 * PS;

  auto load_kv = [&](int kt, int buf) {
    {  // K tile 16x64 -> 256 x b128, one per thread
      int row = tid >> 4, kq = (tid & 15) * 4;
      copy16_g2l(Kg + (size_t)(kt + row) * DK + kq, &Kl[buf][row * KS + kq]);
    }
#pragma unroll
    for (int j = 0; j < 4; j++) {  // V tile 16x256 -> 1024 x b128
      int v = tid + j * 256;
      int row = v >> 6, cq = (v & 63) * 4;
      copy16_g2l(Vg + (size_t)(kt + row) * DVP + cq, &Vl[buf][row * VS + cq]);
    }
  };

  // Q fragments for this wave's 16 rows (A-layout, 16 k-steps over DK=64)
  v2f qf[16];
  {
    const float* qb = qws + ((size_t)bh * Tt + q0 + rg * 16 + ln) * DK;
#pragma unroll
    for (int s = 0; s < 16; s++) {
      int kk = 4 * s + half * 2;
      qf[s].x = qb[kk];
      qf[s].y = qb[kk + 1];
    }
  }

  v8f O[8];
#pragma unroll
  for (int t = 0; t < 8; t++) O[t] = (v8f){0.f, 0.f, 0.f, 0.f, 0.f, 0.f, 0.f, 0.f};
  float mrow[8], lrow[8];
#pragma unroll
  for (int vr = 0; vr < 8; vr++) { mrow[vr] = -1e30f; lrow[vr] = 0.f; }

  const int kmax = q0 + 128;  // OFF + q0 + 64, multiple of KT
  load_kv(0, 0);
  async_join();
  __syncthreads();

  for (int kt = 0; kt < kmax; kt += KT) {
    const int cur = (kt >> 4) & 1;
    if (kt + KT < kmax) load_kv(kt + KT, cur ^ 1);  // prefetch next K/V tile
    const float* Kc = Kl[cur];
    const float* Vc = Vl[cur];

    // scores S (16x16) = Q @ K^T : 16 WMMA k-steps
    v8f S = (v8f){0.f, 0.f, 0.f, 0.f, 0.f, 0.f, 0.f, 0.f};
#pragma unroll
    for (int s = 0; s < 16; s++) {
      int kk = 4 * s + half * 2;
      v2f bf;
      bf.x = Kc[ln * KS + kk];      // B[dk][key] = K[key][dk], n = key = ln
      bf.y = Kc[ln * KS + kk + 1];
      S = wmma4(qf[s], bf, S);
    }

    // scale + causal mask; online softmax
    const int kg = kt + ln;
    float rmax[8], rsum[8];
#pragma unroll
    for (int vr = 0; vr < 8; vr++) {
      int qg = q0 + rg * 16 + vr + half * 8;
      float sv = S[vr] * 0.125f;
      if (kg > qg + OFF) sv = -1e30f;
      S[vr] = sv;
    }
#pragma unroll
    for (int vr = 0; vr < 8; vr++) {
      float x = S[vr];
      x = fmaxf(x, __shfl_xor(x, 1, 32));
      x = fmaxf(x, __shfl_xor(x, 2, 32));
      x = fmaxf(x, __shfl_xor(x, 4, 32));
      x = fmaxf(x, __shfl_xor(x, 8, 32));
      rmax[vr] = x;
    }
    float alpha[8];
#pragma unroll
    for (int vr = 0; vr < 8; vr++) {
      float mn = fmaxf(mrow[vr], rmax[vr]);
      alpha[vr] = __expf(mrow[vr] - mn);
      mrow[vr] = mn;
      S[vr] = __expf(S[vr] - mn);
    }
#pragma unroll
    for (int vr = 0; vr < 8; vr++) {
      float x = S[vr];
      x += __shfl_xor(x, 1, 32);
      x += __shfl_xor(x, 2, 32);
      x += __shfl_xor(x, 4, 32);
      x += __shfl_xor(x, 8, 32);
      rsum[vr] = x;
      lrow[vr] = lrow[vr] * alpha[vr] + rsum[vr];
    }
#pragma unroll
    for (int t = 0; t < 8; t++)
#pragma unroll
      for (int vr = 0; vr < 8; vr++) O[t][vr] = O[t][vr] * alpha[vr];

    // P (C-layout) -> per-wave LDS -> A-layout fragments
#pragma unroll
    for (int vr = 0; vr < 8; vr++) Pw[(vr + half * 8) * PS + ln] = S[vr];
    v2f pa[4];
#pragma unroll
    for (int s = 0; s < 4; s++) {
      int kk = 4 * s + half * 2;
      pa[s].x = Pw[ln * PS + kk];
      pa[s].y = Pw[ln * PS + kk + 1];
    }
    // O += P @ V : 8 col tiles x 4 k-steps
#pragma unroll
    for (int t = 0; t < 8; t++) {
      int c0 = ch * 128 + t * 16;
#pragma unroll
      for (int s = 0; s < 4; s++) {
        int kk = 4 * s + half * 2;
        v2f bf;
        bf.x = Vc[kk * VS + c0 + ln];
        bf.y = Vc[(kk + 1) * VS + c0 + ln];
        O[t] = wmma4(pa[s], bf, O[t]);
      }
    }
    async_join();
    __syncthreads();
  }

  // normalize + write first DV columns, per-head
#pragma unroll
  for (int vr = 0; vr < 8; vr++) lrow[vr] = 1.0f / lrow[vr];
#pragma unroll
  for (int t = 0; t < 8; t++)
#pragma unroll
    for (int vr = 0; vr < 8; vr++) {
      int row = vr + half * 8;
      int col = ch * 128 + t * 16 + ln;
      if (col < DV)
        rotheads[((size_t)bh * Tt + q0 + rg * 16 + row) * DV + col] =
            O[t][vr] * lrow[vr];
    }
}

// ============================================================================
// Kernel 3: per-token Kronecker rotation. One wave per token, 4 waves / block.
// expm via scaling(2^-6)+squaring with Horner Taylor-8, 16x16 matmuls = WMMA.
// Two 8x8 factors packed block-diagonally into one 16x16 expm.
// ============================================================================
#define MS 20
#define XS 68
#define WLDS 2240  // 192 + 3*320 + 1088 floats per wave

__device__ __forceinline__ void matmul16(const float* A, const float* Bm,
                                         float* D, float ascale, bool addI,
                                         int lane) {
  const int half = lane >> 4, ln = lane & 15;
  v2f af[4], bf[4];
#pragma unroll
  for (int s = 0; s < 4; s++) {
    int kk = 4 * s + half * 2;
    af[s].x = A[ln * MS + kk] * ascale;
    af[s].y = A[ln * MS + kk + 1] * ascale;
    bf[s].x = Bm[kk * MS + ln];
    bf[s].y = Bm[(kk + 1) * MS + ln];
  }
  v8f c = (v8f){0.f, 0.f, 0.f, 0.f, 0.f, 0.f, 0.f, 0.f};
#pragma unroll
  for (int s = 0; s < 4; s++) c = wmma4(af[s], bf[s], c);
#pragma unroll
  for (int vr = 0; vr < 8; vr++) {
    int row = vr + half * 8;
    float v = c[vr];
    if (addI && row == ln) v += 1.0f;
    D[row * MS + ln] = v;
  }
}

__device__ __forceinline__ void expm16(float* A, float* P, int lane) {
  // A pre-scaled by 2^-6; result in P.
  for (int e = lane; e < 256; e += 32)
    P[(e >> 4) * MS + (e & 15)] = ((e >> 4) == (e & 15)) ? 1.0f : 0.0f;
#pragma unroll
  for (int j = 8; j >= 1; --j) matmul16(A, P, P, 1.0f / (float)j, true, lane);
#pragma unroll
  for (int t = 0; t < 6; t++) matmul16(P, P, P, 1.0f, false, lane);
}

__global__ __launch_bounds__(128) void rot_kernel(
    const float* __restrict__ x, const float* __restrict__ rotheads,
    float* __restrict__ y) {
  __shared__ __align__(16) float lds[4 * WLDS];
  const int wave = threadIdx.x >> 5, lane = threadIdx.x & 31;
  const int half = lane >> 4, ln = lane & 15;
  const size_t tok = (size_t)blockIdx.x * 4 + wave;  // exactly 4096
  const int b = (int)(tok >> 11), t = (int)(tok & 2047);

  float* prm = lds + wave * WLDS;  // 192 params
  float* A = prm + 192;            // 16xMS
  float* P = A + 320;              // 16xMS
  float* R16 = P + 320;            // 16xMS
  float* X = R16 + 320;            // 16xXS

  // 1) sum rotation params over heads
  for (int d = lane; d < DV; d += 32) {
    float s = 0.f;
#pragma unroll
    for (int h = 0; h < Hh; h++)
      s += rotheads[(((size_t)b * Hh + h) * Tt + t) * DV + d];
    prm[d] = s;
  }

  const float sc = 1.0f / 64.0f;  // 2^-6 pre-scaling for expm

  // 2) R16 = expm(skew(p[0:120]))
  for (int e = lane; e < 256; e += 32) {
    int r = e >> 4, c = e & 15;
    float v = 0.f;
    if (r < c)      v =  prm[r * (31 - r) / 2 + (c - r - 1)] * sc;
    else if (r > c) v = -prm[c * (31 - c) / 2 + (r - c - 1)] * sc;
    A[r * MS + c] = v;
  }
  expm16(A, P, lane);
  for (int e = lane; e < 256; e += 32)
    R16[(e >> 4) * MS + (e & 15)] = P[(e >> 4) * MS + (e & 15)];

  // 3) P = expm(blockdiag(skew(p[120:148]), skew(p[148:176])))
  for (int e = lane; e < 256; e += 32) {
    int r = e >> 4, c = e & 15;
    float v = 0.f;
    if ((r >> 3) == (c >> 3)) {
      int base = 120 + (r >> 3) * 28;
      int lr = r & 7, lc = c & 7;
      if (lr < lc)      v =  prm[base + lr * (15 - lr) / 2 + (lc - lr - 1)] * sc;
      else if (lr > lc) v = -prm[base + lc * (15 - lc) / 2 + (lr - lc - 1)] * sc;
    }
    A[r * MS + c] = v;
  }
  expm16(A, P, lane);

  // 4) load token x as X[16][64]  (a = factor-16 axis, col = b8*8 + c8)
  for (int i = lane; i < 1024; i += 32)
    X[(i >> 6) * XS + (i & 63)] = x[tok * Cc + i];

  // 5) axis-0 rotation: X = R16 @ X  (4 col tiles, in-place, WMMA)
#pragma unroll
  for (int ct = 0; ct < 4; ct++) {
    int c0 = ct * 16;
    v2f af[4], bf[4];
#pragma unroll
    for (int s = 0; s < 4; s++) {
      int kk = 4 * s + half * 2;
      af[s].x = R16[ln * MS + kk];
      af[s].y = R16[ln * MS + kk + 1];
      bf[s].x = X[kk * XS + c0 + ln];
      bf[s].y = X[(kk + 1) * XS + c0 + ln];
    }
    v8f c = (v8f){0.f, 0.f, 0.f, 0.f, 0.f, 0.f, 0.f, 0.f};
#pragma unroll
    for (int s = 0; s < 4; s++) c = wmma4(af[s], bf[s], c);
#pragma unroll
    for (int vr = 0; vr < 8; vr++) X[(vr + half * 8) * XS + c0 + ln] = c[vr];
  }

  // 6) axis-1 rotation: R1 = P[0:8,0:8], contract middle factor (VALU)
  for (int j = lane; j < 128; j += 32) {
    int a = j >> 3, c8 = j & 7;
    float tv[8], ov[8];
#pragma unroll
    for (int bp = 0; bp < 8; bp++) tv[bp] = X[a * XS + bp * 8 + c8];
#pragma unroll
    for (int bo = 0; bo < 8; bo++) {
      float s = 0.f;
#pragma unroll
      for (int bp = 0; bp < 8; bp++) s += P[bo * MS + bp] * tv[bp];
      ov[bo] = s;
    }
#pragma unroll
    for (int bo = 0; bo < 8; bo++) X[a * XS + bo * 8 + c8] = ov[bo];
  }

  // 7) axis-2 rotation: R2 = P[8:16,8:16], contract last factor (VALU)
  for (int j = lane; j < 128; j += 32) {
    int a = j >> 3, b8 = j & 7;
    float tv[8], ov[8];
#pragma unroll
    for (int cp = 0; cp < 8; cp++) tv[cp] = X[a * XS + b8 * 8 + cp];
#pragma unroll
    for (int co = 0; co < 8; co++) {
      float s = 0.f;
#pragma unroll
      for (int cp = 0; cp < 8; cp++) s += P[(8 + co) * MS + 8 + cp] * tv[cp];
      ov[co] = s;
    }
#pragma unroll
    for (int co = 0; co < 8; co++) X[a * XS + b8 * 8 + co] = ov[co];
  }

  // 8) write y
  for (int i = lane; i < 1024; i += 32)
    y[tok * Cc + i] = X[(i >> 6) * XS + (i & 63)];
}

// ============================================================================
// Small init kernels: KV-cache copy + V pad zeroing
// ============================================================================
__global__ void copy_k_kernel(const float* __restrict__ ck, float* __restrict__ kout) {
  size_t i = (size_t)blockIdx.x * 256 + threadIdx.x;  // B*H*OFF*DK
  size_t bh = i / (OFF * DK);
  size_t r = (i / DK) % OFF, d = i % DK;
  kout[(bh * LL + r) * DK + d] = ck[i];
}
__global__ void copy_v_kernel(const float* __restrict__ cv, float* __restrict__ vout) {
  size_t i = (size_t)blockIdx.x * 256 + threadIdx.x;  // B*H*OFF*DVP
  size_t bh = i / (OFF * DVP);
  size_t r = (i / DVP) % OFF, d = i % DVP;
  vout[(bh * LL + r) * DVP + d] = cv[i];
}
__global__ void zero_pad_kernel(float* __restrict__ vout) {
  size_t i = (size_t)blockIdx.x * 256 + threadIdx.x;  // B*H*T*(DVP-DV)
  size_t bh = i / ((size_t)Tt * (DVP - DV));
  size_t t = (i / (DVP - DV)) % Tt, d = DV + i % (DVP - DV);
  vout[(bh * LL + OFF + t) * DVP + d] = 0.f;
}

// ============================================================================
extern "C" void kernel_launch(void* const* d_in, const int* in_sizes, int n_in,
                              void* d_out, int out_size, void* d_ws, size_t ws_size,
                              hipStream_t stream) {
  const float* x = (const float*)d_in[0];
  const float* cached_k = (const float*)d_in[1];
  const float* cached_v = (const float*)d_in[2];
  const float* W_QK = (const float*)d_in[3];
  const float* W_V = (const float*)d_in[4];

  float* y = (float*)d_out;                       // 4,194,304
  float* kout = y + (size_t)Bb * Tt * Cc;         // + 4,325,376
  float* vout = kout + (size_t)Bb * Hh * LL * DK; // + 17,301,504

  float* qws = (float*)d_ws;                              // 16 MB
  float* rotheads = qws + (size_t)Bb * Hh * Tt * DK;      // 46 MB

  copy_k_kernel<<<(Bb * Hh * OFF * DK) / 256, 256, 0, stream>>>(cached_k, kout);
  copy_v_kernel<<<(Bb * Hh * OFF * DVP) / 256, 256, 0, stream>>>(cached_v, vout);
  zero_pad_kernel<<<((size_t)Bb * Hh * Tt * (DVP - DV)) / 256, 256, 0, stream>>>(vout);

  gemm_proj_kernel<<<dim3(32, 32), 256, 0, stream>>>(x, W_QK, 0, qws, kout, vout);
  gemm_proj_kernel<<<dim3(32, 44), 256, 0, stream>>>(x, W_V, 1, qws, kout, vout);

  attn_kernel<<<Bb * Hh * (Tt / 64), 256, 0, stream>>>(qws, kout, vout, rotheads);

  rot_kernel<<<(Bb * Tt) / 4, 128, 0, stream>>>(x, rotheads, y);
}